// MyModel_61933428416342
// MI455X (gfx1250) — compile-verified
//
#include <hip/hip_runtime.h>

// CDNA5 (gfx1250) fused 2-layer LSTM ×2 using V_WMMA_F32_16X16X4_F32.
// One wave32 = one 16-batch tile of one LSTM; loops over T with both layers fused.
//
// WMMA formulation (transposed): D[M=gate, N=batch] = A(W,16x4) x B(x^T/h^T,4x16) + C
//   A layout (32-bit 16x4): lane = 16*half + M ; VGPR0 = (M, K=2*half), VGPR1 = (M, K=2*half+1)
//   B layout (32-bit 4x16): lane = 16*half + N ; VGPR0 = (K=2*half, N), VGPR1 = (K=2*half+1, N)
//   C/D layout:             VGPR v, lane = 16*half + N holds (M = v + 8*half, N)
// M = gate (torch order i,f,g,o ×4). Each lane owns ONLY hidden units j = 2*half+{0,1}
// (exactly what the next-step B operand and the v2f output store need), so the tail is
// 4 cross-half permlanex16 swaps + 2 cell updates per lane — no redundant half.
// Bias is folded into the pre-activation fma (first WMMA uses inline-0 C).
// All activations go through native v_tanh_f32 when available: sigmoid(z) = 0.5+0.5*tanh(z/2).

typedef __attribute__((ext_vector_type(2))) float v2f;
typedef __attribute__((ext_vector_type(8))) float v8f;

#define LSTM_T 2048
#define LSTM_B 4096
#define LSTM_H 4

#if __has_builtin(__builtin_amdgcn_tanhf)
#define TANHF(x) __builtin_amdgcn_tanhf(x)
#else
#define NLOG2E -1.4426950408889634f
__device__ __forceinline__ float tanh_fallback(float z) {
  return __builtin_fmaf(
      2.0f, __builtin_amdgcn_rcpf(1.0f + __builtin_amdgcn_exp2f(z * (2.0f * NLOG2E))),
      -1.0f);
}
#define TANHF(x) tanh_fallback(x)
#endif

// lane <-> lane^16 value exchange as pure VALU (v_permlanex16_b32, identity row selects).
__device__ __forceinline__ float xhalf_swap(float v) {
  unsigned u = __builtin_bit_cast(unsigned, v);
  unsigned r = __builtin_amdgcn_permlanex16(u, u, 0x76543210u, 0xFEDCBA98u, false, false);
  return __builtin_bit_cast(float, r);
}

// One LSTM cell step for a 16-batch tile.
//   Bin/Bh_prev: inputs in WMMA B layout. Kb: bias pre-scaled by each slot's tanh arg coeff.
//   cst: this lane's two cell-state values (hidden j = 2*half+k).
// Returns the new hidden pair — which IS the next-step B-layout operand and the store data.
__device__ __forceinline__ v2f lstm_cell(v2f Bin, v2f Bh_prev,
                                         v2f Awih, v2f Awhh, v8f Kb, v2f& cst,
                                         bool hi, float p, float q, float poff) {
  v8f z0 = {0, 0, 0, 0, 0, 0, 0, 0};  // inline-0 C for the first WMMA
  v8f d = __builtin_amdgcn_wmma_f32_16x16x4_f32(false, Awih, false, Bin,
                                                (short)0, z0, false, false);
  d = __builtin_amdgcn_wmma_f32_16x16x4_f32(false, Awhh, false, Bh_prev,
                                            (short)0, d, false, false);

  float act[8];
  // slots 0..3: gate i (low lanes, sigmoid) / g (high lanes, tanh), branchless:
  //   act = p * tanh(q*z + q*bias) + poff   (low: p=q=0.5, poff=0.5 ; high: p=q=1, poff=0)
#pragma unroll
  for (int v = 0; v < 4; ++v)
    act[v] = __builtin_fmaf(p, TANHF(__builtin_fmaf(d[v], q, Kb[v])), poff);
  // slots 4..7: gate f (low) / o (high) — sigmoid for every lane.
#pragma unroll
  for (int v = 4; v < 8; ++v)
    act[v] = __builtin_fmaf(0.5f, TANHF(__builtin_fmaf(d[v], 0.5f, Kb[v])), 0.5f);

  v2f h;
#pragma unroll
  for (int k = 0; k < 2; ++k) {
    // Ship to the other half exactly what it needs; receive what we need:
    //   low lane j=k   : has i_j,f_j  -> needs g_j,o_j (high slots k, 4+k)
    //   high lane j=2+k: has g_j,o_j  -> needs i_j,f_j (low slots 2+k, 6+k)
    float u  = hi ? act[k]     : act[2 + k];
    float w  = hi ? act[4 + k] : act[6 + k];
    float su = xhalf_swap(u);   // g_j (low) / i_j (high)
    float sw = xhalf_swap(w);   // o_j (low) / f_j (high)
    float myA = hi ? act[2 + k] : act[k];       // i_j (low) / g_j (high)
    float myB = hi ? act[6 + k] : act[4 + k];   // f_j (low) / o_j (high)
    float fv = hi ? sw  : myB;                  // forget gate
    float ov = hi ? myB : sw;                   // output gate
    float cj = __builtin_fmaf(fv, cst[k], myA * su);  // c = f*c + i*g (i*g select-free)
    cst[k] = cj;
    h[k] = ov * TANHF(cj);
  }
  return h;  // = B-layout hidden operand for this lane
}

__global__ __launch_bounds__(32)
void lstm2x2_wmma_kernel(const float* __restrict__ x,
                         const float* __restrict__ w_ih_d, const float* __restrict__ w_hh_d,
                         const float* __restrict__ b_ih_d, const float* __restrict__ b_hh_d,
                         const float* __restrict__ w_ih_n, const float* __restrict__ w_hh_n,
                         const float* __restrict__ b_ih_n, const float* __restrict__ b_hh_n,
                         float* __restrict__ out) {
  const int lane = threadIdx.x;   // 0..31
  const int half = lane >> 4;     // lane half selects K-pair (A/B) and M-half (C/D)
  const int nn   = lane & 15;     // M for A loads, N (= batch-in-tile) for B/C/D
  const bool hi  = (half != 0);

  const int which = blockIdx.y;   // 0 = "_d" LSTM, 1 = "_n" LSTM
  const float* wih = which ? w_ih_n : w_ih_d;
  const float* whh = which ? w_hh_n : w_hh_d;
  const float* bih = which ? b_ih_n : b_ih_d;
  const float* bhh = which ? b_hh_n : b_hh_d;
  float* o = out + (size_t)which * (size_t)LSTM_T * LSTM_B * LSTM_H;

  // Per-lane activation coefficients for slots 0..3 (sigmoid low / tanh high).
  const float q    = hi ? 1.0f : 0.5f;
  const float p    = q;
  const float poff = hi ? 0.0f : 0.5f;

  // Loop-invariant weights in WMMA A layout; combined bias pre-scaled by each slot's
  // tanh-argument coefficient (slots 0..3: q, slots 4..7: 0.5), in C/D layout.
  v2f Awih[2], Awhh[2];
  v8f Kb[2];
#pragma unroll
  for (int l = 0; l < 2; ++l) {
    const float* wi = wih + l * 64;   // [16 gates][4]
    const float* wh = whh + l * 64;   // [16 gates][4]
    Awih[l].x = wi[nn * 4 + 2 * half];
    Awih[l].y = wi[nn * 4 + 2 * half + 1];
    Awhh[l].x = wh[nn * 4 + 2 * half];
    Awhh[l].y = wh[nn * 4 + 2 * half + 1];
#pragma unroll
    for (int v = 0; v < 8; ++v) {
      int g = v + 8 * half;           // M = gate index for this VGPR slot
      float b = bih[l * 16 + g] + bhh[l * 16 + g];
      Kb[l][v] = b * ((v < 4) ? q : 0.5f);
    }
  }

  v2f c1 = {0.0f, 0.0f}, c2 = {0.0f, 0.0f};   // per-lane cell state (hidden 2*half+{0,1})
  v2f Bh1 = {0.0f, 0.0f}, Bh2 = {0.0f, 0.0f}; // hidden state in B layout

  const int b0 = blockIdx.x * 16;                       // batch tile base
  const float* xl = x + (size_t)(b0 + nn) * 4 + 2 * half;
  float*       ol = o + (size_t)(b0 + nn) * 4 + 2 * half;
  const int stride = LSTM_B * 4;                        // floats per timestep

#pragma unroll 1
  for (int t = 0; t < LSTM_T; ++t) {
    v2f Bx = *(const v2f*)xl;                           // x_t^T in B layout (8B/lane, coalesced)
    size_t pf = (t + 8 < LSTM_T) ? (size_t)8 * stride : (size_t)0;
    __builtin_prefetch(xl + pf, 0, 1);                  // global_prefetch_b8, 8 steps ahead

    // Layer 1 consumes x_t; layer 2 consumes layer-1's fresh h (same step) — fully fused.
    v2f Bh1n = lstm_cell(Bx,   Bh1, Awih[0], Awhh[0], Kb[0], c1, hi, p, q, poff);
    v2f Bh2n = lstm_cell(Bh1n, Bh2, Awih[1], Awhh[1], Kb[1], c2, hi, p, q, poff);
    Bh1 = Bh1n;
    Bh2 = Bh2n;

    *(v2f*)ol = Bh2n;  // out[t][b][2*half .. 2*half+1]; 256B contiguous per wave, EXEC all-1s
    xl += stride;
    ol += stride;
  }
}

extern "C" void kernel_launch(void* const* d_in, const int* in_sizes, int n_in,
                              void* d_out, int out_size, void* d_ws, size_t ws_size,
                              hipStream_t stream) {
  (void)in_sizes; (void)n_in; (void)out_size; (void)d_ws; (void)ws_size;
  const float* x      = (const float*)d_in[0];
  const float* w_ih_d = (const float*)d_in[1];
  const float* w_hh_d = (const float*)d_in[2];
  const float* b_ih_d = (const float*)d_in[3];
  const float* b_hh_d = (const float*)d_in[4];
  const float* w_ih_n = (const float*)d_in[5];
  const float* w_hh_n = (const float*)d_in[6];
  const float* b_ih_n = (const float*)d_in[7];
  const float* b_hh_n = (const float*)d_in[8];

  dim3 grid(LSTM_B / 16, 2, 1);   // 256 batch tiles x 2 LSTMs = 512 independent waves
  dim3 block(32, 1, 1);           // one wave32 per block
  lstm2x2_wmma_kernel<<<grid, block, 0, stream>>>(
      x, w_ih_d, w_hh_d, b_ih_d, b_hh_d, w_ih_n, w_hh_n, b_ih_n, b_hh_n,
      (float*)d_out);
}